// VirtualNode_309237645702
// MI455X (gfx1250) — compile-verified
//
#include <hip/hip_runtime.h>

#define NGRAPH 512
#define DIM    256
#define DIM4   64      // DIM / 4 float4s per row
#define SPLIT  8

typedef float v2f __attribute__((ext_vector_type(2)));
typedef float v8f __attribute__((ext_vector_type(8)));

// ---------------------------------------------------------------------------
// Phase 1: segment boundaries. batch_idx is sorted ascending in [0, NGRAPH).
// seg_start[g] = lower_bound(idx, g); seg_start[NGRAPH] = N.
// ---------------------------------------------------------------------------
__global__ void seg_bounds_kernel(const int* __restrict__ idx, int n,
                                  int* __restrict__ seg_start) {
    int g = blockIdx.x * blockDim.x + threadIdx.x;   // 0..NGRAPH inclusive
    if (g > NGRAPH) return;
    int lo = 0, hi = n;
    while (lo < hi) {
        int mid = (lo + hi) >> 1;
        if (idx[mid] < g) lo = mid + 1; else hi = mid;
    }
    seg_start[g] = lo;
}

// ---------------------------------------------------------------------------
// Phase 2: deterministic split-K segment partial sums.
// grid = (NGRAPH, SPLIT); block = 64 threads, one float4 column-group each.
// 4 independent accumulators keep 4 row-loads (4 KB/block) in flight.
// ---------------------------------------------------------------------------
__global__ void seg_partial_kernel(const float* __restrict__ h,
                                   const int* __restrict__ seg_start,
                                   float* __restrict__ partial) {
    const int g = blockIdx.x;
    const int s = blockIdx.y;
    const int lo = seg_start[g];
    const int hi = seg_start[g + 1];
    const int len = hi - lo;
    const int per = (len + SPLIT - 1) / SPLIT;
    int r0 = lo + s * per;
    int r1 = r0 + per; if (r1 > hi) r1 = hi;
    const int c = threadIdx.x;                       // 0..63
    const float4* __restrict__ h4 = (const float4*)h;

    float4 a0 = {0.f,0.f,0.f,0.f}, a1 = a0, a2 = a0, a3 = a0;
    int r = r0;
    for (; r + 4 <= r1; r += 4) {
        float4 x0 = h4[(long long)(r + 0) * DIM4 + c];
        float4 x1 = h4[(long long)(r + 1) * DIM4 + c];
        float4 x2 = h4[(long long)(r + 2) * DIM4 + c];
        float4 x3 = h4[(long long)(r + 3) * DIM4 + c];
        a0.x += x0.x; a0.y += x0.y; a0.z += x0.z; a0.w += x0.w;
        a1.x += x1.x; a1.y += x1.y; a1.z += x1.z; a1.w += x1.w;
        a2.x += x2.x; a2.y += x2.y; a2.z += x2.z; a2.w += x2.w;
        a3.x += x3.x; a3.y += x3.y; a3.z += x3.z; a3.w += x3.w;
    }
    for (; r < r1; ++r) {
        float4 x = h4[(long long)r * DIM4 + c];
        a0.x += x.x; a0.y += x.y; a0.z += x.z; a0.w += x.w;
    }
    float4 acc;
    acc.x = (a0.x + a1.x) + (a2.x + a3.x);
    acc.y = (a0.y + a1.y) + (a2.y + a3.y);
    acc.z = (a0.z + a1.z) + (a2.z + a3.z);
    acc.w = (a0.w + a1.w) + (a2.w + a3.w);
    ((float4*)partial)[((long long)s * NGRAPH + g) * DIM4 + c] = acc;
}

// ---------------------------------------------------------------------------
// Phase 3: fixed-order reduction of the SPLIT partials -> S[NGRAPH, DIM].
// ---------------------------------------------------------------------------
__global__ void seg_reduce_kernel(const float* __restrict__ partial,
                                  float* __restrict__ S) {
    int i = blockIdx.x * blockDim.x + threadIdx.x;   // over NGRAPH*DIM4
    if (i >= NGRAPH * DIM4) return;
    const float4* __restrict__ p4 = (const float4*)partial;
    float4 acc = p4[i];
    #pragma unroll
    for (int s = 1; s < SPLIT; ++s) {
        float4 t = p4[(long long)s * NGRAPH * DIM4 + i];
        acc.x += t.x; acc.y += t.y; acc.z += t.z; acc.w += t.w;
    }
    ((float4*)S)[i] = acc;
}

// ---------------------------------------------------------------------------
// Phase 4: h_G = relu(S @ W + b) via fp32 WMMA 16x16x4.
// One wave per 16x16 output tile. 512 tiles = 64 blocks * 8 waves.
// A layout (16x4 f32): lane m = lane&15, VGPR v holds K = v + 2*(lane>>4)
// B layout (4x16 f32): lane n = lane&15, VGPR v holds K = v + 2*(lane>>4)
// C/D layout (16x16 f32): lane n = lane&15, VGPR v holds M = v + 8*(lane>>4)
// ---------------------------------------------------------------------------
__global__ void gemm_relu_kernel(const float* __restrict__ S,
                                 const float* __restrict__ W,
                                 const float* __restrict__ bias,
                                 float* __restrict__ hG) {
    const int wave  = (blockIdx.x * blockDim.x + threadIdx.x) >> 5;
    const int lane  = threadIdx.x & 31;
    const int tileM = wave >> 4;          // 0..31  (512 graphs / 16)
    const int tileN = wave & 15;          // 0..15  (256 feats / 16)
    const int half  = lane >> 4;          // 0 or 1
    const int t16   = lane & 15;
    const int rowA  = tileM * 16 + t16;   // graph row this lane supplies for A
    const int colB  = tileN * 16 + t16;   // output column this lane holds

    v8f c = {};
    #pragma unroll 4
    for (int k0 = 0; k0 < DIM; k0 += 4) {
        const int ka = k0 + half * 2;
        v2f a, b;
        a.x = S[rowA * DIM + ka];
        a.y = S[rowA * DIM + ka + 1];
        b.x = W[ka * DIM + colB];
        b.y = W[(ka + 1) * DIM + colB];
        c = __builtin_amdgcn_wmma_f32_16x16x4_f32(
                /*neg_a=*/false, a, /*neg_b=*/false, b,
                /*c_mod=*/(short)0, c, /*reuse_a=*/false, /*reuse_b=*/false);
    }

    const float bb = bias[colB];
    #pragma unroll
    for (int v = 0; v < 8; ++v) {
        const int m = tileM * 16 + v + half * 8;
        float val = c[v] + bb;
        hG[m * DIM + colB] = val > 0.f ? val : 0.f;
    }
}

// ---------------------------------------------------------------------------
// Phase 5: out[n] = hG[idx[n]] + h[n]   (dominant 410 MB stream, float4)
// ---------------------------------------------------------------------------
__global__ void gather_add_kernel(const float* __restrict__ h,
                                  const int* __restrict__ idx,
                                  const float* __restrict__ hG,
                                  float* __restrict__ out, int n) {
    long long i = (long long)blockIdx.x * blockDim.x + threadIdx.x;
    long long total = (long long)n * DIM4;
    if (i >= total) return;
    int row = (int)(i >> 6);              // node
    int c   = (int)(i & (DIM4 - 1));      // float4 column
    int g   = idx[row];
    float4 a  = ((const float4*)h)[i];
    float4 bv = ((const float4*)hG)[(long long)g * DIM4 + c];
    float4 o;
    o.x = a.x + bv.x; o.y = a.y + bv.y; o.z = a.z + bv.z; o.w = a.w + bv.w;
    ((float4*)out)[i] = o;
}

// ---------------------------------------------------------------------------
extern "C" void kernel_launch(void* const* d_in, const int* in_sizes, int n_in,
                              void* d_out, int out_size, void* d_ws, size_t ws_size,
                              hipStream_t stream) {
    const float* h   = (const float*)d_in[0];
    const int*   idx = (const int*)  d_in[1];
    const float* W   = (const float*)d_in[2];
    const float* b   = (const float*)d_in[3];
    float* out = (float*)d_out;
    const int N = in_sizes[1];            // number of nodes

    // workspace layout (all 16B aligned)
    float* partial = (float*)d_ws;                        // SPLIT*NGRAPH*DIM
    float* S       = partial + (size_t)SPLIT * NGRAPH * DIM;  // NGRAPH*DIM
    float* hG      = S + (size_t)NGRAPH * DIM;                // NGRAPH*DIM
    int*   seg     = (int*)(hG + (size_t)NGRAPH * DIM);       // NGRAPH+1

    // 1) segment boundaries
    seg_bounds_kernel<<<(NGRAPH + 1 + 63) / 64, 64, 0, stream>>>(idx, N, seg);

    // 2) deterministic split partial segment sums
    dim3 gridB(NGRAPH, SPLIT);
    seg_partial_kernel<<<gridB, 64, 0, stream>>>(h, seg, partial);

    // 3) reduce partials
    seg_reduce_kernel<<<(NGRAPH * DIM4 + 255) / 256, 256, 0, stream>>>(partial, S);

    // 4) WMMA fp32 GEMM + bias + ReLU
    gemm_relu_kernel<<<64, 256, 0, stream>>>(S, W, b, hG);

    // 5) gather + residual
    long long total = (long long)N * DIM4;
    int blocks = (int)((total + 255) / 256);
    gather_add_kernel<<<blocks, 256, 0, stream>>>(h, idx, hG, out, N);
}